// VectorQuantizer_89120571392488
// MI455X (gfx1250) — compile-verified
//
#include <hip/hip_runtime.h>
#include <math.h>

// Problem constants (match reference)
#define B_ 16
#define T_ 2048
#define D_ 512
#define K_ 4096
#define N_ (B_ * T_)          // 32768 rows
#define KT_ (K_ / 16)         // 256 code tiles
#define LDSTRIDE 516          // padded LDS row stride (floats): 516 % 64 == 4 -> conflict-free b64 operand loads

typedef __attribute__((ext_vector_type(2))) float v2f;
typedef __attribute__((ext_vector_type(8))) float v8f;
typedef __attribute__((ext_vector_type(4))) unsigned int v4u;
typedef __attribute__((ext_vector_type(4))) int v4i;
typedef __attribute__((ext_vector_type(8))) int v8i;

// ---------------------------------------------------------------------------
// Kernel 1: codebook squared norms  wnorm[k] = sum_d W[k][d]^2
// ---------------------------------------------------------------------------
__global__ __launch_bounds__(256) void vq_wnorm(const float* __restrict__ W,
                                                float* __restrict__ wnorm) {
    const int wave = threadIdx.x >> 5;
    const int lane = threadIdx.x & 31;
    const int k = blockIdx.x * 8 + wave;
    const float4* row = (const float4*)(W + (size_t)k * D_);
    float s = 0.0f;
#pragma unroll
    for (int i = 0; i < 4; ++i) {
        float4 v = row[lane * 4 + i];
        s += v.x * v.x + v.y * v.y + v.z * v.z + v.w * v.w;
    }
#pragma unroll
    for (int off = 16; off > 0; off >>= 1) s += __shfl_xor(s, off, 32);
    if (lane == 0) wnorm[k] = s;
}

// ---------------------------------------------------------------------------
// Kernel 2: WMMA distance + argmin.
// Block = 256 threads = 8 waves; each wave owns one 16-row M-tile (128 rows).
// X tile staged by the Tensor Data Mover (TDM) with D# LDS padding of
// 2 DWORDs every 256 DWORDs -> row stride 516 with a +2 shift at d>=256.
// W tile prefetched into 8 *named* float4 registers (no alloca -> no scratch
// spill) so global latency overlaps the WMMA loop; published to LDS between
// barriers. __launch_bounds__(256,2): LDS caps us at 1 block/WGP = 2 waves/EU,
// so grant the matching (large) VGPR budget.
// ---------------------------------------------------------------------------
__global__ __launch_bounds__(256, 2)
void vq_argmin(const float* __restrict__ X,
               const float* __restrict__ W,
               const float* __restrict__ wnorm,
               int* __restrict__ idx) {
    extern __shared__ float lds[];
    float* lx = lds;                      // 128 * LDSTRIDE floats (TDM-staged, mid-pad at 256)
    float* lw = lds + 128 * LDSTRIDE;     // 16  * LDSTRIDE floats

    const int tid  = threadIdx.x;
    const int wave = tid >> 5;
    const int lane = tid & 31;
    const int rowblk = blockIdx.x * 128;

    // --- Issue TDM load of the 128x512 f32 X tile into LDS (wave 0 only) ---
    if (tid < 32) {
        unsigned long long ga =
            (unsigned long long)(const void*)(X + (size_t)rowblk * D_);
        // D# group0: count=1 | lds_addr=0 | global_addr | type=2
        v4u g0 = { 1u, 0u, (unsigned)ga,
                   (unsigned)((ga >> 32) & 0x01FFFFFFu) | (2u << 30) };
        // D# group1: data_size=4B, pad_enable, pad_interval=256DW, pad_amount=2DW,
        //            tensor_dim0=512, tensor_dim1=N_, tile_dim0=512, tile_dim1=128,
        //            tensor_dim0_stride=512
        v8i g1 = { (int)((2u << 16) | (1u << 20) | (7u << 22) | (1u << 25)),
                   (int)(512u << 16),            // tensor_dim0 low16 @ [31:16]
                   (int)(0x8000u << 16),         // tensor_dim1 (32768) low16 @ [31:16]
                   (int)(512u << 16),            // tile_dim0 @ [31:16]
                   128,                          // tile_dim1 = 128 rows
                   512,                          // tensor_dim0_stride
                   0, 0 };
        v4i z4 = { 0, 0, 0, 0 };
        v8i z8 = { 0, 0, 0, 0, 0, 0, 0, 0 };
        __builtin_amdgcn_tensor_load_to_lds(g0, g1, z4, z4, z8, 0);
    }

    // --- W tile staging: named registers, flat addressing ---
    // f_i = tid + 256*i  =>  c4 = tid&127 (constant over i), r_i = (tid>>7) + 2*i
    const int r0 = tid >> 7;          // 0 or 1
    const int c4 = tid & 127;
    float4 w0, w1, w2, w3, w4, w5, w6, w7;

#define LOADW(kt2)                                                            \
    {                                                                         \
        const float4* g = (const float4*)(W + (size_t)(kt2) * 16 * D_)        \
                          + (size_t)r0 * 128 + c4;                            \
        w0 = g[0 * 256]; w1 = g[1 * 256]; w2 = g[2 * 256]; w3 = g[3 * 256];   \
        w4 = g[4 * 256]; w5 = g[5 * 256]; w6 = g[6 * 256]; w7 = g[7 * 256];   \
    }
#define STOREW()                                                              \
    {                                                                         \
        float* l = &lw[r0 * LDSTRIDE + c4 * 4];                               \
        *(float4*)(l + 0 * 2 * LDSTRIDE)  = w0;                               \
        *(float4*)(l + 1 * 2 * LDSTRIDE)  = w1;                               \
        *(float4*)(l + 2 * 2 * LDSTRIDE)  = w2;                               \
        *(float4*)(l + 3 * 2 * LDSTRIDE)  = w3;                               \
        *(float4*)(l + 4 * 2 * LDSTRIDE)  = w4;                               \
        *(float4*)(l + 5 * 2 * LDSTRIDE)  = w5;                               \
        *(float4*)(l + 6 * 2 * LDSTRIDE)  = w6;                               \
        *(float4*)(l + 7 * 2 * LDSTRIDE)  = w7;                               \
    }

    LOADW(0);
    STOREW();
    if (tid < 32) __builtin_amdgcn_s_wait_tensorcnt(0);
    __syncthreads();

    const int m  = lane & 15;   // C column / A row index
    const int hi = lane >> 4;   // half-wave K sub-pair select
    const float* ax = &lx[(wave * 16 + m) * LDSTRIDE + 2 * hi];
    const float* bx = &lw[m * LDSTRIDE + 2 * hi];

    float best[8];
    int   bestk[8];
#pragma unroll
    for (int j = 0; j < 8; ++j) { best[j] = 3.4e38f; bestk[j] = 0; }

    for (int kt = 0; kt < KT_; ++kt) {
        if (kt + 1 < KT_) LOADW(kt + 1);        // prefetch next W tile to regs
        const int   kidx = kt * 16 + m;
        const float wn   = wnorm[kidx];          // hoisted: hidden under WMMA loop

        v8f acc0 = {}, acc1 = {}, acc2 = {}, acc3 = {};
        // d in [0,256): TDM pad not yet applied
#pragma unroll 4
        for (int d = 0; d < 256; d += 16) {
            v2f a0 = *(const v2f*)(ax + d),      b0 = *(const v2f*)(bx + d);
            v2f a1 = *(const v2f*)(ax + d + 4),  b1 = *(const v2f*)(bx + d + 4);
            v2f a2 = *(const v2f*)(ax + d + 8),  b2 = *(const v2f*)(bx + d + 8);
            v2f a3 = *(const v2f*)(ax + d + 12), b3 = *(const v2f*)(bx + d + 12);
            acc0 = __builtin_amdgcn_wmma_f32_16x16x4_f32(false, a0, false, b0, (short)0, acc0, false, false);
            acc1 = __builtin_amdgcn_wmma_f32_16x16x4_f32(false, a1, false, b1, (short)0, acc1, false, false);
            acc2 = __builtin_amdgcn_wmma_f32_16x16x4_f32(false, a2, false, b2, (short)0, acc2, false, false);
            acc3 = __builtin_amdgcn_wmma_f32_16x16x4_f32(false, a3, false, b3, (short)0, acc3, false, false);
        }
        // d in [256,512): X shifted +2 floats by the TDM mid-row pad
        const float* axp = ax + 2;
#pragma unroll 4
        for (int d = 256; d < 512; d += 16) {
            v2f a0 = *(const v2f*)(axp + d),      b0 = *(const v2f*)(bx + d);
            v2f a1 = *(const v2f*)(axp + d + 4),  b1 = *(const v2f*)(bx + d + 4);
            v2f a2 = *(const v2f*)(axp + d + 8),  b2 = *(const v2f*)(bx + d + 8);
            v2f a3 = *(const v2f*)(axp + d + 12), b3 = *(const v2f*)(bx + d + 12);
            acc0 = __builtin_amdgcn_wmma_f32_16x16x4_f32(false, a0, false, b0, (short)0, acc0, false, false);
            acc1 = __builtin_amdgcn_wmma_f32_16x16x4_f32(false, a1, false, b1, (short)0, acc1, false, false);
            acc2 = __builtin_amdgcn_wmma_f32_16x16x4_f32(false, a2, false, b2, (short)0, acc2, false, false);
            acc3 = __builtin_amdgcn_wmma_f32_16x16x4_f32(false, a3, false, b3, (short)0, acc3, false, false);
        }

        v8f acc = (acc0 + acc1) + (acc2 + acc3);
#pragma unroll
        for (int j = 0; j < 8; ++j) {
            float s = wn - 2.0f * acc[j];
            if (s < best[j]) { best[j] = s; bestk[j] = kidx; }
        }

        __syncthreads();                         // all waves done reading tile kt
        if (kt + 1 < KT_) STOREW();              // publish next tile
        __syncthreads();
    }
#undef LOADW
#undef STOREW

    // C layout: element (j, lane) = C[m = j + 8*hi][n = lane&15].
    // Min-reduce each row across the 16 lanes of its half (first-index ties).
#pragma unroll
    for (int j = 0; j < 8; ++j) {
        float s = best[j];
        int   k = bestk[j];
#pragma unroll
        for (int off = 1; off < 16; off <<= 1) {
            float os = __shfl_xor(s, off, 32);
            int   ok = __shfl_xor(k, off, 32);
            if (os < s || (os == s && ok < k)) { s = os; k = ok; }
        }
        best[j] = s; bestk[j] = k;
    }
    if (lane == 0 || lane == 16) {
        int rbase = rowblk + wave * 16 + (lane >> 1);   // lane 16 -> +8
#pragma unroll
        for (int j = 0; j < 8; ++j) idx[rbase + j] = bestk[j];
    }
}

// ---------------------------------------------------------------------------
// Kernel 3: gather quantized rows, write straight-through output, accumulate
// masked SSE and codebook histogram. One wave per row (8 rows/block).
// ---------------------------------------------------------------------------
__global__ __launch_bounds__(256) void vq_quantize(const float* __restrict__ X,
                                                   const float* __restrict__ W,
                                                   const int* __restrict__ idx,
                                                   const int* __restrict__ dlen,
                                                   float* __restrict__ outq,
                                                   float* __restrict__ sse_acc,
                                                   int* __restrict__ counts) {
    const int wave = threadIdx.x >> 5;
    const int lane = threadIdx.x & 31;
    const int n = blockIdx.x * 8 + wave;
    const int k = idx[n];

    const float* xr = X + (size_t)n * D_;
    const float* wr = W + (size_t)k * D_;
    float* qr = outq + (size_t)n * D_;   // d_out+1 base -> coalesced b32 stores

    float sse = 0.0f;
#pragma unroll
    for (int i = 0; i < 16; ++i) {
        int c = i * 32 + lane;           // coalesced
        float x = xr[c];
        float q = wr[c];
        float dqx = q - x;
        sse += dqx * dqx;
        qr[c] = x + dqx;                 // exact straight-through arithmetic
    }
#pragma unroll
    for (int off = 16; off > 0; off >>= 1) sse += __shfl_xor(sse, off, 32);
    if (lane == 0) {
        atomicAdd(&counts[k], 1);
        int b = n / T_;
        int t = n % T_;
        if (2 * t < dlen[b])             // t < data_len/2
            atomicAdd(sse_acc, sse * (1.0f / (float)D_));
    }
}

// Scatter the one-hot ones (encodings zeroed by hipMemsetAsync).
__global__ void vq_scatter(float* __restrict__ enc, const int* __restrict__ idx) {
    int n = blockIdx.x * 256 + threadIdx.x;
    enc[(size_t)n * K_ + idx[n]] = 1.0f;
}

// ---------------------------------------------------------------------------
// Kernel 4: finalize loss and perplexity (single block).
// ---------------------------------------------------------------------------
__global__ __launch_bounds__(256) void vq_finalize(const int* __restrict__ counts,
                                                   const float* __restrict__ sse_acc,
                                                   const int* __restrict__ dlen,
                                                   float* __restrict__ loss_out,
                                                   float* __restrict__ perp_out) {
    __shared__ float red[256];
    float s = 0.0f;
    for (int k = threadIdx.x; k < K_; k += 256) {
        float p = (float)counts[k] * (1.0f / (float)N_);
        s += p * __logf(p + 1e-10f);
    }
    red[threadIdx.x] = s;
    __syncthreads();
    for (int off = 128; off > 0; off >>= 1) {
        if (threadIdx.x < off) red[threadIdx.x] += red[threadIdx.x + off];
        __syncthreads();
    }
    if (threadIdx.x == 0) {
        long nv = 0;
        for (int b = 0; b < B_; ++b) {
            int dl = dlen[b];
            long c = (long)((dl + 1) / 2);   // #{t : 2t < dl}
            if (c > T_) c = T_;
            nv += c;
        }
        *loss_out = 1.25f * (*sse_acc) / (float)nv;
        *perp_out = __expf(-red[0]);
    }
}

// ---------------------------------------------------------------------------
// Launch: out = [loss | quantized_st (N*D) | perplexity | encodings (N*K)]
// ---------------------------------------------------------------------------
extern "C" void kernel_launch(void* const* d_in, const int* in_sizes, int n_in,
                              void* d_out, int out_size, void* d_ws, size_t ws_size,
                              hipStream_t stream) {
    const float* X    = (const float*)d_in[0];   // [B,T,D] f32
    const float* W    = (const float*)d_in[1];   // [K,D]   f32
    const int*   dlen = (const int*)d_in[2];     // [B]     int

    float* out      = (float*)d_out;
    float* loss_out = out;
    float* q_out    = out + 1;
    float* perp_out = out + 1 + (size_t)N_ * D_;
    float* enc      = out + 2 + (size_t)N_ * D_;

    char*  ws     = (char*)d_ws;
    int*   counts = (int*)ws;                          // K ints
    float* sse    = (float*)(ws + (size_t)K_ * 4);     // 1 float
    int*   idx    = (int*)(ws + 16512);                // N ints
    float* wnorm  = (float*)(ws + 16512 + (size_t)N_ * 4);  // K floats

    // Zero histogram + SSE accumulator, and the one-hot encodings region.
    (void)hipMemsetAsync(ws, 0, (size_t)K_ * 4 + 4, stream);
    (void)hipMemsetAsync(enc, 0, (size_t)N_ * K_ * sizeof(float), stream);

    vq_wnorm<<<K_ / 8, 256, 0, stream>>>(W, wnorm);

    const size_t ldsbytes = (size_t)(128 + 16) * LDSTRIDE * sizeof(float); // ~290 KB
    vq_argmin<<<N_ / 128, 256, ldsbytes, stream>>>(X, W, wnorm, idx);

    vq_quantize<<<N_ / 8, 256, 0, stream>>>(X, W, idx, dlen, q_out, sse, counts);
    vq_scatter<<<N_ / 256, 256, 0, stream>>>(enc, idx);

    vq_finalize<<<1, 256, 0, stream>>>(counts, sse, dlen, loss_out, perp_out);
}